// DetectionIOUMetric_36807869727204
// MI455X (gfx1250) — compile-verified
//
#include <hip/hip_runtime.h>
#include <math.h>
#include <stdint.h>

#define PBOX 4096
#define GBOX 1024
#define CCH 6
#define NTHREADS 256
#define NSLICES 4                    // 4 pred-quarters per batch -> 128 blocks
#define PSLICE (PBOX / NSLICES)      // 1024 preds per block
#define TILE 512                     // preds per LDS tile in pass B
#define NTILES (PSLICE / TILE)       // 2 tiles per block
#define EPSF 1e-7f
#define NEG_INF_F (-3.402823466e38f)
#define MAP_NEG_INF 0x007FFFFFu      // order-map of -inf
#define MAP_HALF    0xBF000000u      // order-map of 0.5f

// ---------------- CDNA5 async global->LDS helpers (ASYNCcnt path) ------------
__device__ __forceinline__ void async_b128(uint32_t lds_off, const void* gaddr) {
    asm volatile("global_load_async_to_lds_b128 %0, %1, off"
                 :: "v"(lds_off), "v"(gaddr)
                 : "memory");
}
__device__ __forceinline__ void wait_async0() {
    asm volatile("s_wait_asynccnt 0x0" ::: "memory");
}
__device__ __forceinline__ uint32_t lds_off(const void* p) {
    return (uint32_t)(uintptr_t)p;   // low 32 bits of flat shared addr = LDS offset
}
// Monotone order-preserving float->uint map (strictly increasing)
__device__ __forceinline__ unsigned fmap(float f) {
    unsigned u = __float_as_uint(f);
    return (u & 0x80000000u) ? ~u : (u | 0x80000000u);
}

// ---------------------------------------------------------------------------
__global__ void init_ws(unsigned* __restrict__ gmax, int ngmax,
                        unsigned* __restrict__ cnt, int ncnt) {
    int i = blockIdx.x * blockDim.x + threadIdx.x;
    if (i < ngmax) gmax[i] = MAP_NEG_INF;
    if (i < ncnt)  cnt[i]  = 0u;
}

__global__ __launch_bounds__(NTHREADS) void iou_main(
        const float* __restrict__ pred,   // [B, PBOX, 6]
        const float* __restrict__ gt,     // [B, GBOX, 6]
        unsigned* __restrict__ gmax,      // [B, GBOX] mapped partial maxima
        unsigned* __restrict__ cnt)       // [B][3] = pos, npred, ngt
{
    const int b     = blockIdx.y;
    const int slice = blockIdx.x;              // 0..NSLICES-1
    const int tid   = threadIdx.x;

    __shared__ float4 s_gtc[GBOX];             // x1,y1,x2,y2     (16 KB)
    __shared__ float2 s_gta[GBOX];             // area, validf    ( 8 KB)
    __shared__ __align__(16) float s_stage[2][TILE * CCH];   // 2 x 12 KB
    __shared__ unsigned s_red[4];

    // ---- Stage all gt boxes (24 KB spans both stage buffers) via async DMA
    {
        const char* src = (const char*)(gt + (size_t)b * GBOX * CCH);
        const int nchunks = GBOX * CCH / 4;    // 1536 x 16B
        for (int i = tid; i < nchunks; i += NTHREADS)
            async_b128(lds_off(&s_stage[0][0] + i * 4), src + (size_t)i * 16);
        wait_async0();
        __syncthreads();
    }

    // ---- Derive packed gt corners/area/valid
    unsigned my_gt_valid = 0;
    for (int g = tid; g < GBOX; g += NTHREADS) {
        const float* q = &s_stage[0][0] + g * CCH;
        float cx = q[0], cy = q[1], w = q[2], h = q[3];
        float hx = 0.5f * w, hy = 0.5f * h;
        float x1 = cx - hx, y1 = cy - hy, x2 = cx + hx, y2 = cy + hy;
        s_gtc[g] = make_float4(x1, y1, x2, y2);
        float vf = (cx != -1.0f) ? 1.0f : 0.0f;
        s_gta[g] = make_float2((x2 - x1) * (y2 - y1), vf);
        my_gt_valid += (cx != -1.0f) ? 1u : 0u;
    }
    __syncthreads();

    // ---- Prefetch this block's pred tile 0 (overlaps with pass A)
    {
        const char* src =
            (const char*)(pred + ((size_t)b * PBOX + (size_t)slice * PSLICE) * CCH);
        const int nchunks = TILE * CCH / 4;    // 768 x 16B
        for (int i = tid; i < nchunks; i += NTHREADS)
            async_b128(lds_off(&s_stage[0][0] + i * 4), src + (size_t)i * 16);
    }

    // ====== Pass A: pred-major, 4 preds/thread register-blocked =============
    float px1[4], py1[4], px2[4], py2[4], pa[4], mxv[4];
    bool  pv[4];
    unsigned my_pred_valid = 0;
    #pragma unroll
    for (int k = 0; k < 4; ++k) {
        int p = slice * PSLICE + k * NTHREADS + tid;
        const float* pp = pred + ((size_t)b * PBOX + p) * CCH;
        float cx = pp[0], cy = pp[1], w = pp[2], h = pp[3];
        float hx = 0.5f * w, hy = 0.5f * h;
        px1[k] = cx - hx; py1[k] = cy - hy; px2[k] = cx + hx; py2[k] = cy + hy;
        pa[k]  = (px2[k] - px1[k]) * (py2[k] - py1[k]);
        pv[k]  = (cx != -1.0f);
        mxv[k] = NEG_INF_F;
        my_pred_valid += pv[k] ? 1u : 0u;
    }
    #pragma unroll 2
    for (int g = 0; g < GBOX; ++g) {
        float4 c = s_gtc[g];                 // ds_load_b128 broadcast
        float2 a = s_gta[g];                 // ds_load_b64 broadcast
        bool gv = (a.y != 0.0f);
        #pragma unroll
        for (int k = 0; k < 4; ++k) {
            float ix1 = fmaxf(px1[k], c.x), iy1 = fmaxf(py1[k], c.y);
            float ix2 = fminf(px2[k], c.z), iy2 = fminf(py2[k], c.w);
            float iw = fmaxf(ix2 - ix1, 0.0f), ih = fmaxf(iy2 - iy1, 0.0f);
            float inter = iw * ih;
            float uni = pa[k] + a.x - inter;
            float iou = inter * __builtin_amdgcn_rcpf(uni + EPSF);  // v_rcp_f32
            float m = (pv[k] || gv) ? iou : 0.0f;
            mxv[k] = fmaxf(mxv[k], m);
        }
    }
    unsigned my_pos = 0;
    #pragma unroll
    for (int k = 0; k < 4; ++k) my_pos += (mxv[k] > 0.5f) ? 1u : 0u;

    // ====== Pass B: gt-major, 4 gts/thread over this block's pred quarter ===
    float4 gc[4]; float2 ga[4]; float mg[4];
    #pragma unroll
    for (int k = 0; k < 4; ++k) {
        gc[k] = s_gtc[tid * 4 + k];
        ga[k] = s_gta[tid * 4 + k];
        mg[k] = NEG_INF_F;
    }
    for (int t = 0; t < NTILES; ++t) {
        const int cur = t & 1;
        wait_async0();          // my async chunks of tile t landed
        __syncthreads();        // everyone's chunks landed; prev buffer free
        if (t + 1 < NTILES) {   // stream next tile into the other buffer
            const char* src = (const char*)(pred +
                ((size_t)b * PBOX + (size_t)slice * PSLICE + (size_t)(t + 1) * TILE) * CCH);
            const int nchunks = TILE * CCH / 4;
            for (int i = tid; i < nchunks; i += NTHREADS)
                async_b128(lds_off(&s_stage[cur ^ 1][0] + i * 4), src + (size_t)i * 16);
        }
        const float* st = &s_stage[cur][0];
        #pragma unroll 2
        for (int p = 0; p < TILE; ++p) {
            float cx = st[p * CCH + 0], cy = st[p * CCH + 1];
            float w  = st[p * CCH + 2], h  = st[p * CCH + 3];
            float hx = 0.5f * w, hy = 0.5f * h;
            float qx1 = cx - hx, qy1 = cy - hy, qx2 = cx + hx, qy2 = cy + hy;
            float areap = (qx2 - qx1) * (qy2 - qy1);
            bool validp = (cx != -1.0f);
            #pragma unroll
            for (int k = 0; k < 4; ++k) {
                float ix1 = fmaxf(qx1, gc[k].x), iy1 = fmaxf(qy1, gc[k].y);
                float ix2 = fminf(qx2, gc[k].z), iy2 = fminf(qy2, gc[k].w);
                float iw = fmaxf(ix2 - ix1, 0.0f), ih = fmaxf(iy2 - iy1, 0.0f);
                float inter = iw * ih;
                float uni = areap + ga[k].x - inter;
                float iou = inter * __builtin_amdgcn_rcpf(uni + EPSF);
                float m = (validp || (ga[k].y != 0.0f)) ? iou : 0.0f;
                mg[k] = fmaxf(mg[k], m);
            }
        }
    }
    // Deterministic cross-block combine: order-mapped u32 atomic max
    #pragma unroll
    for (int k = 0; k < 4; ++k)
        atomicMax(&gmax[(size_t)b * GBOX + tid * 4 + k], fmap(mg[k]));

    // ---- Block-reduce counts, one global atomic per counter per block
    if (tid < 4) s_red[tid] = 0u;
    __syncthreads();
    atomicAdd(&s_red[0], my_pos);
    atomicAdd(&s_red[1], my_pred_valid);
    atomicAdd(&s_red[2], my_gt_valid);
    __syncthreads();
    if (tid == 0) {
        atomicAdd(&cnt[b * 3 + 0], s_red[0]);
        atomicAdd(&cnt[b * 3 + 1], s_red[1]);
        if (slice == 0) atomicAdd(&cnt[b * 3 + 2], s_red[2]);  // gt counted once
    }
}

__global__ __launch_bounds__(NTHREADS) void finalize(
        const unsigned* __restrict__ gmax,  // [B, GBOX]
        const unsigned* __restrict__ cnt,   // [B][3]
        float* __restrict__ out, int B)
{
    const int b = blockIdx.x;
    const int tid = threadIdx.x;
    __shared__ unsigned s_true;
    if (tid == 0) s_true = 0u;
    __syncthreads();
    unsigned local_true = 0;
    for (int g = tid; g < GBOX; g += NTHREADS)
        local_true += (gmax[(size_t)b * GBOX + g] > MAP_HALF) ? 1u : 0u;
    atomicAdd(&s_true, local_true);
    __syncthreads();
    if (tid == 0) {
        float np  = (float)cnt[b * 3 + 0];
        float npr = (float)cnt[b * 3 + 1];
        float ng  = (float)cnt[b * 3 + 2];
        float nt  = (float)s_true;
        float prec = np / (npr + EPSF);
        float rec  = nt / (ng + EPSF);
        float f = 2.0f * (prec * rec) / (prec + rec + EPSF);
        out[b]         = prec;
        out[B + b]     = rec;
        out[2 * B + b] = f;
    }
}

extern "C" void kernel_launch(void* const* d_in, const int* in_sizes, int n_in,
                              void* d_out, int out_size, void* d_ws, size_t ws_size,
                              hipStream_t stream) {
    (void)n_in; (void)out_size; (void)ws_size;
    const float* pred = (const float*)d_in[0];   // [B, 4096, 6] f32
    const float* gt   = (const float*)d_in[1];   // [B, 1024, 6] f32
    float* out = (float*)d_out;                  // 3*B floats (prec, rec, f)
    const int B = in_sizes[0] / (PBOX * CCH);

    unsigned* gmax = (unsigned*)d_ws;            // B*GBOX mapped maxima
    unsigned* cnt  = gmax + (size_t)B * GBOX;    // B*3 counters

    const int ninit = B * GBOX;
    init_ws<<<(ninit + NTHREADS - 1) / NTHREADS, NTHREADS, 0, stream>>>(
        gmax, ninit, cnt, B * 3);
    dim3 grid(NSLICES, B);
    iou_main<<<grid, NTHREADS, 0, stream>>>(pred, gt, gmax, cnt);
    finalize<<<B, NTHREADS, 0, stream>>>(gmax, cnt, out, B);
}